// Fourier_20469814133246
// MI455X (gfx1250) — compile-verified
//
#include <hip/hip_runtime.h>
#include <math.h>

// ============================================================================
// Band-limited 3D inverse rFFT, spectral support |k| <= 4 (257 real DOF).
// out[n0,n1,n2] = (1/N) * sum_{|k|<=4} F(k) exp(2pi i (k0 n0/32 + k1 n1/128 + k2 n2/128))
// Separable pipeline:
//   stage1 (kernel A): H[k0=0..4][k1=-4..4][n2] = sum_k2 F * e^{i 2pi k2 n2/128}
//   stage2 (kernel B): G[k0][n1,n2]            = sum_k1 H * e^{i 2pi k1 n1/128}
//   stage3 (kernel B): out = W[32x12] * Gr[12x16] tiles via
//                      V_WMMA_F32_16X16X4_F32 (fp32 matrix core path).
// Write-bandwidth bound: 256 MB out / 23.3 TB/s ~= 11 us floor. Output uses
// non-temporal stores so the 5.9 MB H intermediate stays resident in L2.
// Workspace: H = 128 samples * 5*9*128 * 8B = 5,898,240 bytes.
// ============================================================================

typedef __attribute__((ext_vector_type(2))) float v2f;
typedef __attribute__((ext_vector_type(8))) float v8f;

#define TWO_PI 6.28318530717958647692f
#define DIM 257

// feed predicate: position (a0,b0,k2) of the 9x9x5 candidate box is a stored
// (non-Hermitian-overwritten) half-spectrum coefficient inside the k-ball.
// a0: 0..4 -> k0=0..4 (i=0..4), 5..8 -> k0=-4..-1 (i=28..31). Same for b0/k1.
__device__ __forceinline__ bool feed_re(int a0, int b0, int k2) {
    int k0 = (a0 <= 4) ? a0 : a0 - 9;
    int k1 = (b0 <= 4) ? b0 : b0 - 9;
    if (k0 * k0 + k1 * k1 + k2 * k2 > 16) return false;
    if (k2 == 0) {
        if (a0 > 4) return false;             // i>16: Hermitian-filled row
        if (a0 == 0 && b0 > 4) return false;  // i==0, j>64: Hermitian-filled
    }
    return true;
}

// ---------------------------------------------------------------------------
// Kernel A: scatter x -> small spectrum in LDS, then k2-transform -> H (ws).
// grid = 128 samples, block = 128 threads (one thread per n2).
// H layout: H[((sample*5 + k0)*9 + b0)*128 + n2]  (float2)
// ---------------------------------------------------------------------------
__global__ void __launch_bounds__(128)
fourier_stage1(const float* __restrict__ x, float2* __restrict__ H) {
    const int sample = blockIdx.x;
    const int tid = threadIdx.x;

    __shared__ float Sre[9][9][5];
    __shared__ float Sim[9][9][5];
    for (int p = tid; p < 9 * 9 * 5; p += 128) {
        (&Sre[0][0][0])[p] = 0.0f;
        (&Sim[0][0][0])[p] = 0.0f;
    }
    __syncthreads();

    // Rebuild FEED_IDX ordering on device: feed positions in flat (i,j,k)
    // order, all real parts first, then all imag parts.
    const float* xs = x + sample * DIM;
    for (int c = tid; c < 405; c += 128) {
        int a0 = c / 45, r = c % 45, b0 = r / 5, k2 = r % 5;
        int rank_re = 0, rank_im = 0, nre = 0;
        for (int cc = 0; cc < 405; ++cc) {
            int aa = cc / 45, rr = cc % 45, bb = rr / 5, kk = rr % 5;
            bool fre = feed_re(aa, bb, kk);
            bool fim = fre && !(aa == 0 && bb == 0 && kk == 0);
            if (cc < c) { rank_re += fre ? 1 : 0; rank_im += fim ? 1 : 0; }
            nre += fre ? 1 : 0;
        }
        bool fre = feed_re(a0, b0, k2);
        bool fim = fre && !(a0 == 0 && b0 == 0 && k2 == 0);
        if (fre) Sre[a0][b0][k2] = xs[rank_re];
        if (fim) Sim[a0][b0][k2] = xs[nre + rank_im];
    }
    __syncthreads();

    // k2 transform: H(k0,k1,n2) = sum_{k2=-4..4} F e^{i 2pi k2 n2/128},
    // negative k2 via F(k0,k1,-k2) = conj(F(-k0,-k1,k2)). Only k0>=0 stored.
    const int n2 = tid;
    float c_[5], s_[5];
    c_[0] = 1.0f; s_[0] = 0.0f;
    #pragma unroll
    for (int k2 = 1; k2 <= 4; ++k2) {
        float ang = TWO_PI * (float)(k2 * n2) * (1.0f / 128.0f);
        __sincosf(ang, &s_[k2], &c_[k2]);
    }
    for (int k0p = 0; k0p < 5; ++k0p) {
        int na = (k0p == 0) ? 0 : 9 - k0p;   // index of -k0
        for (int b0 = 0; b0 < 9; ++b0) {
            int nb = (b0 == 0) ? 0 : 9 - b0; // index of -k1
            float hr, hi;
            if (k0p == 0 && b0 > 4) {        // k2==0 Hermitian-filled entry
                hr = Sre[0][nb][0]; hi = -Sim[0][nb][0];
            } else {
                hr = Sre[k0p][b0][0]; hi = Sim[k0p][b0][0];
            }
            #pragma unroll
            for (int k2 = 1; k2 <= 4; ++k2) {
                float cc = c_[k2], ss = s_[k2];
                float fr = Sre[k0p][b0][k2], fi = Sim[k0p][b0][k2];
                hr += fr * cc - fi * ss;     // +k2 term
                hi += fr * ss + fi * cc;
                float gr = Sre[na][nb][k2], gi = -Sim[na][nb][k2];
                hr += gr * cc + gi * ss;     // -k2 term (conj * e^{-i})
                hi += gi * cc - gr * ss;
            }
            H[((sample * 5 + k0p) * 9 + b0) * 128 + n2] = make_float2(hr, hi);
        }
    }
}

// ---------------------------------------------------------------------------
// Kernel B: k1 transform (per-lane VALU, uniform twiddles per block) + k0
// transform as WMMA f32 16x16x4 GEMM. grid = (n1=128, sample=128),
// block = 256 threads = 8 waves, wave w owns n2 in [16w, 16w+16).
// ---------------------------------------------------------------------------
__global__ void __launch_bounds__(256)
fourier_stage23(const float2* __restrict__ H, float* __restrict__ out) {
    const int n1 = blockIdx.x;
    const int sample = blockIdx.y;
    const int tid = threadIdx.x;
    const int lane = tid & 31;
    const int wave = tid >> 5;
    const int l16 = lane & 15;
    const bool hi = lane >= 16;
    const int n2 = wave * 16 + l16;   // lanes 16..31 mirror 0..15 (B layout)

    // twiddles e^{i 2pi k1 n1/128} in b0 storage order (uniform per block)
    float twr[9], twi[9];
    twr[0] = 1.0f; twi[0] = 0.0f;
    #pragma unroll
    for (int k = 1; k <= 4; ++k) {
        float ang = TWO_PI * (float)(k * n1) * (1.0f / 128.0f);
        float s, c; __sincosf(ang, &s, &c);
        twr[k] = c;     twi[k] = s;      // b0=k   -> k1=+k
        twr[9 - k] = c; twi[9 - k] = -s; // b0=9-k -> k1=-k
    }

    // stage 2: Gr[12] = [G0.re, G1.re, G1.im, ..., G4.re, G4.im, 0, 0, 0]
    float Gr[12];
    const float2* Hs = H + (size_t)sample * (5 * 9 * 128);
    #pragma unroll
    for (int k0p = 0; k0p < 5; ++k0p) {
        float gr = 0.0f, gi = 0.0f;
        #pragma unroll
        for (int b0 = 0; b0 < 9; ++b0) {
            float2 h = Hs[(k0p * 9 + b0) * 128 + n2];
            gr += h.x * twr[b0] - h.y * twi[b0];
            gi += h.x * twi[b0] + h.y * twr[b0];
        }
        if (k0p == 0) { Gr[0] = gr; }
        else { Gr[2 * k0p - 1] = gr; Gr[2 * k0p] = gi; }
    }
    Gr[9] = 0.0f; Gr[10] = 0.0f; Gr[11] = 0.0f;

    // stage-3 weights, branch-free (no EXEC-divergent regions between WMMAs):
    // W[m][0] = invN; W[m][2kk-1] = 2invN*cos(2pi kk m/32);
    // W[m][2kk] = -2invN*sin(...); W[m][9..11] = 0.
    const float invN = 1.0f / 524288.0f;
    const float twoInvN = 2.0f * invN;
    float cw0[4], sw0[4], cw1[4], sw1[4];   // rows m=l16 and m=l16+16
    #pragma unroll
    for (int kk = 1; kk <= 4; ++kk) {
        float s, c;
        __sincosf(TWO_PI * (float)(kk * l16) * (1.0f / 32.0f), &s, &c);
        cw0[kk - 1] = twoInvN * c; sw0[kk - 1] = -twoInvN * s;
        __sincosf(TWO_PI * (float)(kk * (l16 + 16)) * (1.0f / 32.0f), &s, &c);
        cw1[kk - 1] = twoInvN * c; sw1[kk - 1] = -twoInvN * s;
    }

    // stage 3: out_tile[32 x 16] = W[32 x 12] * Gr[12 x 16]
    // V_WMMA_F32_16X16X4_F32 layouts (ISA 7.12.2):
    //  A 16x4:  lane l: M=l&15; vgpr0 = K=(l<16?0:2)+kb, vgpr1 = K+1
    //  B 4x16:  lane l: N=l&15; vgpr0 = K=(l<16?0:2)+kb row, vgpr1 = K+1
    //  C/D:     vgpr r: M = r + (l<16?0:8), N = l&15
    v8f d0 = {0.f, 0.f, 0.f, 0.f, 0.f, 0.f, 0.f, 0.f};
    v8f d1 = {0.f, 0.f, 0.f, 0.f, 0.f, 0.f, 0.f, 0.f};
    v2f a0v, a1v, bv;

    // kb = 0: lo lanes K={0,1} -> {invN, cw[0]}; hi lanes K={2,3} -> {sw[0], cw[1]}
    bv[0]  = hi ? Gr[2]  : Gr[0];  bv[1]  = hi ? Gr[3]  : Gr[1];
    a0v[0] = hi ? sw0[0] : invN;   a0v[1] = hi ? cw0[1] : cw0[0];
    a1v[0] = hi ? sw1[0] : invN;   a1v[1] = hi ? cw1[1] : cw1[0];
    d0 = __builtin_amdgcn_wmma_f32_16x16x4_f32(false, a0v, false, bv, (short)0, d0, false, false);
    d1 = __builtin_amdgcn_wmma_f32_16x16x4_f32(false, a1v, false, bv, (short)0, d1, false, false);

    // kb = 4: lo K={4,5} -> {sw[1], cw[2]}; hi K={6,7} -> {sw[2], cw[3]}
    bv[0]  = hi ? Gr[6]  : Gr[4];  bv[1]  = hi ? Gr[7]  : Gr[5];
    a0v[0] = hi ? sw0[2] : sw0[1]; a0v[1] = hi ? cw0[3] : cw0[2];
    a1v[0] = hi ? sw1[2] : sw1[1]; a1v[1] = hi ? cw1[3] : cw1[2];
    d0 = __builtin_amdgcn_wmma_f32_16x16x4_f32(false, a0v, false, bv, (short)0, d0, false, false);
    d1 = __builtin_amdgcn_wmma_f32_16x16x4_f32(false, a1v, false, bv, (short)0, d1, false, false);

    // kb = 8: lo K={8,9} -> {sw[3], 0}; hi K={10,11} -> {0, 0}
    bv[0]  = hi ? Gr[10] : Gr[8];  bv[1]  = hi ? Gr[11] : Gr[9];
    a0v[0] = hi ? 0.0f   : sw0[3]; a0v[1] = 0.0f;
    a1v[0] = hi ? 0.0f   : sw1[3]; a1v[1] = 0.0f;
    d0 = __builtin_amdgcn_wmma_f32_16x16x4_f32(false, a0v, false, bv, (short)0, d0, false, false);
    d1 = __builtin_amdgcn_wmma_f32_16x16x4_f32(false, a1v, false, bv, (short)0, d1, false, false);

    // store: out[((sample*32 + n0)*128 + n1)*128 + n2], non-temporal (stream-
    // ing 256 MB; keep H resident in L2 instead).
    const size_t base = ((size_t)(sample * 32) * 128 + (size_t)n1) * 128 + (size_t)n2;
    const size_t n0stride = 128 * 128;
    #pragma unroll
    for (int r = 0; r < 8; ++r) {
        int n0a = r + (hi ? 8 : 0);
        __builtin_nontemporal_store(d0[r], &out[base + (size_t)n0a * n0stride]);
        __builtin_nontemporal_store(d1[r], &out[base + (size_t)(n0a + 16) * n0stride]);
    }
}

// ---------------------------------------------------------------------------
extern "C" void kernel_launch(void* const* d_in, const int* in_sizes, int n_in,
                              void* d_out, int out_size, void* d_ws, size_t ws_size,
                              hipStream_t stream) {
    (void)in_sizes; (void)n_in; (void)out_size; (void)ws_size;
    const float* x = (const float*)d_in[0];     // [128 samples][257]
    float* out = (float*)d_out;                 // [128][32][128][128]
    float2* H = (float2*)d_ws;                  // 5,898,240 bytes used

    fourier_stage1<<<dim3(128), dim3(128), 0, stream>>>(x, H);
    fourier_stage23<<<dim3(128, 128), dim3(256), 0, stream>>>(H, out);
}